// LightGCN_30846455120739
// MI455X (gfx1250) — compile-verified
//
#include <hip/hip_runtime.h>
#include <hip/hip_bf16.h>
#include <math.h>

typedef float v2f __attribute__((ext_vector_type(2)));
typedef float v8f __attribute__((ext_vector_type(8)));

#define N_NODES   100000
#define N_EDGES   1600000
#define F_IN      64
#define N_CLASSES 47

// ---------------------------------------------------------------------------
// Utility: zero a float buffer, float4-vectorized (n must be multiple of 4)
// ---------------------------------------------------------------------------
__global__ void zero_f32x4_kernel(float4* __restrict__ p, int n4) {
    int i = blockIdx.x * blockDim.x + threadIdx.x;
    if (i < n4) p[i] = make_float4(0.f, 0.f, 0.f, 0.f);
}

// ---------------------------------------------------------------------------
// In-degree at dst via f32 atomics (deg values are small exact integers)
// ---------------------------------------------------------------------------
__global__ void deg_kernel(const int* __restrict__ dst, float* __restrict__ deg, int E) {
    int e = blockIdx.x * blockDim.x + threadIdx.x;
    if (e < E) atomicAdd(&deg[dst[e]], 1.0f);
}

// deg -> dinv in place: deg>0 ? rsqrt(deg) : 0
__global__ void dinv_kernel(float* __restrict__ dinv, int n) {
    int i = blockIdx.x * blockDim.x + threadIdx.x;
    if (i < n) {
        float d = dinv[i];
        dinv[i] = (d > 0.0f) ? rsqrtf(d) : 0.0f;
    }
}

// Precompute per-edge symmetric norm once (shared by both convolutions):
// norm[e] = dinv[src[e]] * dinv[dst[e]]
__global__ void norm_kernel(const int* __restrict__ src, const int* __restrict__ dst,
                            const float* __restrict__ dinv, float* __restrict__ norm,
                            int E) {
    int e = blockIdx.x * blockDim.x + threadIdx.x;
    if (e < E) norm[e] = dinv[src[e]] * dinv[dst[e]];
}

// ---------------------------------------------------------------------------
// LGConv scatter: out[dst] += X[src] * norm[e]
// One thread per (edge, 4 features): one b128 gather + 4 f32 atomics.
// Working set (x 25.6MB, edges 12.8MB, norm 6.4MB, out 25.6MB) is L2-resident
// on the 192MB L2, so this stage is L2-bandwidth / atomic-throughput bound;
// per-word f32 atomics into L2 are the fast path (avg degree ~16 -> heavy
// line reuse, no privatization needed).
// ---------------------------------------------------------------------------
__global__ void conv_scatter_kernel(const float4* __restrict__ X4,
                                    const int* __restrict__ src,
                                    const int* __restrict__ dst,
                                    const float* __restrict__ norm,
                                    float* __restrict__ out, int E) {
    long long idx = (long long)blockIdx.x * blockDim.x + threadIdx.x;
    int e = (int)(idx >> 4);       // 16 threads per edge
    int q = (int)(idx & 15);       // float4 slot: features [4q, 4q+3]
    if (e >= E) return;
    int s = src[e];
    int d = dst[e];
    float nrm = norm[e];
    float4 v = X4[(size_t)s * (F_IN / 4) + q];
    float* o = out + (size_t)d * F_IN + 4 * q;
    atomicAdd(o + 0, v.x * nrm);
    atomicAdd(o + 1, v.y * nrm);
    atomicAdd(o + 2, v.z * nrm);
    atomicAdd(o + 3, v.w * nrm);
}

// ---------------------------------------------------------------------------
// Dense layer via V_WMMA_F32_16X16X4_F32 (exact fp32 path on CDNA5).
// out[M x NOUT] = act( relu(A[M x 64]) @ W[64 x NOUT] + bias )
// One wave computes a 16-row tile across all NOUT columns.
//   A 16x4 f32 frag:  lane = M (mod 16), {half,j} -> K = 2*half + j
//   B  4x16 f32 frag: lane = N (mod 16), {half,j} -> K = 2*half + j
//   C/D 16x16 f32:    lane = N, VGPR r -> M = r + 8*half
// M (=100000) is a multiple of 16; tiles guarded per-wave (uniform branch),
// so EXEC is all-ones inside the WMMA region as the ISA requires.
// ---------------------------------------------------------------------------
template <int NOUT, bool RELU_OUT>
__global__ void lin_wmma_kernel(const float* __restrict__ A,
                                const float* __restrict__ W,
                                const float* __restrict__ bias,
                                float* __restrict__ out, int M) {
    const int wave  = (blockIdx.x * blockDim.x + threadIdx.x) >> 5;
    const int lane  = threadIdx.x & 31;
    const int tileM = wave;
    if (tileM * 16 >= M) return;              // uniform per wave

    const int half = lane >> 4;               // 0 or 1
    const int l16  = lane & 15;
    const int row  = tileM * 16 + l16;        // A row owned by this lane

    constexpr int NT = (NOUT + 15) / 16;
    v8f acc[NT] = {};

    for (int s = 0; s < 16; ++s) {            // K loop: 16 steps of K=4
        const int k0 = 4 * s + 2 * half;      // even -> 8B aligned b64 load
        v2f a = *(const v2f*)&A[(size_t)row * F_IN + k0];
        a.x = fmaxf(a.x, 0.0f);               // fused ReLU(conv output)
        a.y = fmaxf(a.y, 0.0f);
#pragma unroll
        for (int t = 0; t < NT; ++t) {
            const int col = t * 16 + l16;
            v2f b;
            if ((NOUT % 16 == 0) || (col < NOUT)) {
                b.x = W[(size_t)k0 * NOUT + col];
                b.y = W[(size_t)(k0 + 1) * NOUT + col];
            } else {
                b.x = 0.0f; b.y = 0.0f;
            }
            acc[t] = __builtin_amdgcn_wmma_f32_16x16x4_f32(
                false, a, false, b, (short)0, acc[t], false, false);
        }
    }

#pragma unroll
    for (int t = 0; t < NT; ++t) {
#pragma unroll
        for (int r = 0; r < 8; ++r) {
            const int m = tileM * 16 + r + 8 * half;
            const int n = t * 16 + l16;
            if ((NOUT % 16 == 0) || (n < NOUT)) {
                float v = acc[t][r] + bias[n];
                if (RELU_OUT) v = fmaxf(v, 0.0f);
                out[(size_t)m * NOUT + n] = v;
            }
        }
    }
}

// ---------------------------------------------------------------------------
// In-place row log_softmax over 47 classes; one wave32 per row.
// Lane holds classes {lane, lane+32}; shuffle reductions for max / sum.
// ---------------------------------------------------------------------------
__global__ void logsoftmax_kernel(float* __restrict__ out, int M) {
    const int wave = (blockIdx.x * blockDim.x + threadIdx.x) >> 5;
    const int lane = threadIdx.x & 31;
    if (wave >= M) return;                    // uniform per wave
    float* row = out + (size_t)wave * N_CLASSES;

    float v0 = (lane < N_CLASSES) ? row[lane] : -INFINITY;
    float v1 = (lane + 32 < N_CLASSES) ? row[lane + 32] : -INFINITY;

    float m = fmaxf(v0, v1);
#pragma unroll
    for (int off = 16; off > 0; off >>= 1)
        m = fmaxf(m, __shfl_xor(m, off, 32));

    float s = expf(v0 - m) + expf(v1 - m);    // exp(-inf - m) == 0
#pragma unroll
    for (int off = 16; off > 0; off >>= 1)
        s += __shfl_xor(s, off, 32);

    const float lse = m + logf(s);
    if (lane < N_CLASSES)      row[lane]      = v0 - lse;
    if (lane + 32 < N_CLASSES) row[lane + 32] = v1 - lse;
}

// ---------------------------------------------------------------------------
// Launch
// ---------------------------------------------------------------------------
extern "C" void kernel_launch(void* const* d_in, const int* in_sizes, int n_in,
                              void* d_out, int out_size, void* d_ws, size_t ws_size,
                              hipStream_t stream) {
    const float* x    = (const float*)d_in[0];
    const int*   eidx = (const int*)d_in[1];     // [2, E] row-major
    const float* W1   = (const float*)d_in[2];   // [64, 64]
    const float* b1   = (const float*)d_in[3];
    const float* W2   = (const float*)d_in[4];   // [64, 47]
    const float* b2   = (const float*)d_in[5];
    float*       out  = (float*)d_out;           // [N, 47] log-softmax

    const int* src = eidx;
    const int* dst = eidx + N_EDGES;

    // Workspace layout (floats): dinv | norm[E] | bufA (N x 64) | bufB (N x 64)
    float* ws   = (float*)d_ws;
    float* dinv = ws;                            // N floats (deg, then dinv)
    float* norm = ws + 102400;                   // E floats
    float* bufA = norm + 1638400;                // padded past norm
    float* bufB = bufA + (size_t)N_NODES * F_IN;

    const int T = 256;
    const int nodeBlocks = (N_NODES + T - 1) / T;
    const int featN      = N_NODES * F_IN;
    const int edgeBlocks = (N_EDGES + T - 1) / T;
    const long long ef4  = (long long)N_EDGES * (F_IN / 4);      // 16 thr/edge
    const int convBlocks = (int)((ef4 + T - 1) / T);
    const int nodeZ4     = ((N_NODES + T * 4 - 1) / (T * 4));    // zero dinv (pad ok)
    const int featZ4     = (featN / 4 + T - 1) / T;
    const int tileWaves  = N_NODES / 16;                         // 6250, exact
    const int linBlocks  = (tileWaves + 8 - 1) / 8;              // 8 waves / block
    const int smBlocks   = (N_NODES + 8 - 1) / 8;                // 8 rows / block

    // Degree + symmetric norm (norm[e] shared by both convolutions)
    zero_f32x4_kernel<<<nodeZ4, T, 0, stream>>>((float4*)dinv, N_NODES / 4);
    deg_kernel<<<edgeBlocks, T, 0, stream>>>(dst, dinv, N_EDGES);
    dinv_kernel<<<nodeBlocks, T, 0, stream>>>(dinv, N_NODES);
    norm_kernel<<<edgeBlocks, T, 0, stream>>>(src, dst, dinv, norm, N_EDGES);

    // conv1: bufA = LGConv(x)
    zero_f32x4_kernel<<<featZ4, T, 0, stream>>>((float4*)bufA, featN / 4);
    conv_scatter_kernel<<<convBlocks, T, 0, stream>>>((const float4*)x, src, dst,
                                                      norm, bufA, N_EDGES);

    // lin1: bufB = relu(relu(bufA) @ W1 + b1)
    lin_wmma_kernel<64, true><<<linBlocks, T, 0, stream>>>(bufA, W1, b1, bufB, N_NODES);

    // conv2: bufA = LGConv(bufB)
    zero_f32x4_kernel<<<featZ4, T, 0, stream>>>((float4*)bufA, featN / 4);
    conv_scatter_kernel<<<convBlocks, T, 0, stream>>>((const float4*)bufB, src, dst,
                                                      norm, bufA, N_EDGES);

    // lin2: out = relu(bufA) @ W2 + b2   (logits)
    lin_wmma_kernel<47, false><<<linBlocks, T, 0, stream>>>(bufA, W2, b2, out, N_NODES);

    // log_softmax in place
    logsoftmax_kernel<<<smBlocks, T, 0, stream>>>(out, N_NODES);
}